// ModelNew_17411797418176
// MI455X (gfx1250) — compile-verified
//
#include <hip/hip_runtime.h>

typedef float v4f __attribute__((ext_vector_type(4)));

#define TPB   256
#define NCOLS 4096
#define VPT   (NCOLS / (TPB * 4))   // 4 float4 chunks per thread -> 16 floats/thread
#define NWAVES (TPB / 32)

__device__ __forceinline__ float wave_sum32(float v) {
#pragma unroll
  for (int m = 16; m; m >>= 1) v += __shfl_xor(v, m, 32);
  return v;
}

__device__ __forceinline__ float wave_max32(float v) {
#pragma unroll
  for (int m = 16; m; m >>= 1) v = fmaxf(v, __shfl_xor(v, m, 32));
  return v;
}

__device__ __forceinline__ float vmax4_abs(v4f a) {
  return fmaxf(fmaxf(fabsf(a.x), fabsf(a.y)), fmaxf(fabsf(a.z), fabsf(a.w)));
}

__device__ __forceinline__ float quant1(float y, float c, float inv_scale) {
  // clip(round_half_even(y*c/scale), -128, 127), stored as float
  return fminf(fmaxf(rintf(y * c * inv_scale), -128.0f), 127.0f);
}

__global__ __launch_bounds__(TPB) void fused_add_rmsnorm_dualquant(
    const float* __restrict__ x1, const float* __restrict__ x2,
    const float* __restrict__ gamma, const float* __restrict__ ss1,
    const float* __restrict__ ss2,
    float* __restrict__ o_sum, float* __restrict__ o_norm,
    float* __restrict__ o_y1, float* __restrict__ o_s1,
    float* __restrict__ o_y2, float* __restrict__ o_s2) {
  const int tid  = threadIdx.x;
  const int wid  = tid >> 5;        // wave32: 8 waves per block
  const int lane = tid & 31;
  const size_t base = (size_t)blockIdx.x * (size_t)NCOLS;

  __shared__ float s_red[NWAVES];
  __shared__ float s_m1[NWAVES];
  __shared__ float s_m2[NWAVES];
  __shared__ float s_bc[3];         // inv_rms, inv_scale1, inv_scale2

  const v4f* p1 = (const v4f*)(x1 + base);
  const v4f* p2 = (const v4f*)(x2 + base);
  const v4f* pg  = (const v4f*)gamma;
  const v4f* pc1 = (const v4f*)ss1;
  const v4f* pc2 = (const v4f*)ss2;
  v4f* po_sum  = (v4f*)(o_sum  + base);
  v4f* po_norm = (v4f*)(o_norm + base);
  v4f* po_y1   = (v4f*)(o_y1   + base);
  v4f* po_y2   = (v4f*)(o_y2   + base);

  // ---- Phase 1: residual add, stream x_sum, accumulate sum of squares ----
  v4f s[VPT];          // row data, register-resident for the whole kernel
  v4f g[VPT];          // gamma (L2-resident regular loads)
  float acc = 0.0f;
#pragma unroll
  for (int i = 0; i < VPT; ++i) {
    const int idx = i * TPB + tid;                    // coalesced b128
    v4f a = __builtin_nontemporal_load(p1 + idx);     // streamed: read-once
    v4f b = __builtin_nontemporal_load(p2 + idx);
    g[i] = pg[idx];
    v4f t = a + b;
    s[i] = t;
    __builtin_nontemporal_store(t, po_sum + idx);     // streamed: write-once
    acc += t.x * t.x + t.y * t.y + t.z * t.z + t.w * t.w;
  }

  acc = wave_sum32(acc);
  if (lane == 0) s_red[wid] = acc;
  __syncthreads();
  if (wid == 0) {
    float v = (lane < NWAVES) ? s_red[lane] : 0.0f;
    v = wave_sum32(v);
    if (lane == 0)
      s_bc[0] = __frsqrt_rn(v * (1.0f / (float)NCOLS) + 1e-5f);
  }
  __syncthreads();
  const float inv_rms = s_bc[0];

  // ---- Phase 2: normalize, stream y_norm, accumulate per-row abs-maxes ----
  v4f c1[VPT], c2[VPT];
  float m1 = 0.0f, m2 = 0.0f;
#pragma unroll
  for (int i = 0; i < VPT; ++i) {
    const int idx = i * TPB + tid;
    v4f y = s[i] * inv_rms * g[i];
    s[i] = y;                                         // keep y in registers
    __builtin_nontemporal_store(y, po_norm + idx);
    c1[i] = pc1[idx];
    c2[i] = pc2[idx];
    m1 = fmaxf(m1, vmax4_abs(y * c1[i]));
    m2 = fmaxf(m2, vmax4_abs(y * c2[i]));
  }

  m1 = wave_max32(m1);
  m2 = wave_max32(m2);
  if (lane == 0) { s_m1[wid] = m1; s_m2[wid] = m2; }
  __syncthreads();
  if (wid == 0) {
    float a = (lane < NWAVES) ? s_m1[lane] : 0.0f;
    float b = (lane < NWAVES) ? s_m2[lane] : 0.0f;
    a = wave_max32(a);
    b = wave_max32(b);
    if (lane == 0) {
      o_s1[blockIdx.x] = a * (1.0f / 127.0f);
      o_s2[blockIdx.x] = b * (1.0f / 127.0f);
      s_bc[1] = 127.0f / a;
      s_bc[2] = 127.0f / b;
    }
  }
  __syncthreads();
  const float i1 = s_bc[1];
  const float i2 = s_bc[2];

  // ---- Phase 3: dual quantize from registers, stream both outputs ----
#pragma unroll
  for (int i = 0; i < VPT; ++i) {
    const int idx = i * TPB + tid;
    const v4f y = s[i];
    v4f q1, q2;
    q1.x = quant1(y.x, c1[i].x, i1); q1.y = quant1(y.y, c1[i].y, i1);
    q1.z = quant1(y.z, c1[i].z, i1); q1.w = quant1(y.w, c1[i].w, i1);
    q2.x = quant1(y.x, c2[i].x, i2); q2.y = quant1(y.y, c2[i].y, i2);
    q2.z = quant1(y.z, c2[i].z, i2); q2.w = quant1(y.w, c2[i].w, i2);
    __builtin_nontemporal_store(q1, po_y1 + idx);
    __builtin_nontemporal_store(q2, po_y2 + idx);
  }
}

extern "C" void kernel_launch(void* const* d_in, const int* in_sizes, int n_in,
                              void* d_out, int out_size, void* d_ws, size_t ws_size,
                              hipStream_t stream) {
  const float* x1    = (const float*)d_in[0];
  const float* x2    = (const float*)d_in[1];
  const float* gamma = (const float*)d_in[2];
  const float* ss1   = (const float*)d_in[3];
  const float* ss2   = (const float*)d_in[4];

  const size_t M = (size_t)in_sizes[0];      // B*S*N elements
  const size_t R = M / (size_t)NCOLS;        // B*S rows

  float* out = (float*)d_out;
  // Tuple layout, flat concat in return order:
  // x_sum[M], y_norm[M], y1[M], scale1[R], y2[M], scale2[R]
  float* o_sum  = out;
  float* o_norm = out + M;
  float* o_y1   = out + 2 * M;
  float* o_s1   = out + 3 * M;
  float* o_y2   = out + 3 * M + R;
  float* o_s2   = out + 4 * M + R;

  fused_add_rmsnorm_dualquant<<<dim3((unsigned)R), dim3(TPB), 0, stream>>>(
      x1, x2, gamma, ss1, ss2, o_sum, o_norm, o_y1, o_s1, o_y2, o_s2);
}